// SoftDTW_7533372637335
// MI455X (gfx1250) — compile-verified
//
#include <hip/hip_runtime.h>
#include <hip/hip_bf16.h>

// ---------------------------------------------------------------------------
// SoftDTW on MI455X (gfx1250):
//   Kernel 1: D = x2 + y2 - 2*X@Y^T via V_WMMA_F32_16X16X4_F32 (full f32 WMMA)
//   Kernel 2: anti-diagonal DP, 1 workgroup per batch, rolling diagonals in LDS
// ---------------------------------------------------------------------------

#define Bsz 64
#define Nsz 512
#define Msz 512
#define DIMsz 64
#define BIGF 1.0e10f

typedef float v2f __attribute__((ext_vector_type(2)));
typedef float v8f __attribute__((ext_vector_type(8)));

// ------------------------- Phase 1: pairwise dist --------------------------
// grid = (B, N/16, M/(16*8)), block = 256 (8 wave32s). Each wave computes one
// 16x16 tile of D for batch b using 16x V_WMMA_F32_16X16X4_F32 over DIM=64.
//
// A (16x4, f32) lane layout: lanes 0-15 hold M=0..15; VGPR0 = K=0 (lo half) /
// K=2 (hi half); VGPR1 = K=1 / K=3.  B (4x16) mirrors with N across lanes.
// C/D (16x16 f32): VGPR v holds row v (lanes 0-15) / row v+8 (lanes 16-31),
// col = lane & 15.
__global__ __launch_bounds__(256)
void softdtw_pairdist_wmma(const float* __restrict__ X,
                           const float* __restrict__ Y,
                           float* __restrict__ D) {
    const int b    = blockIdx.x;
    const int n0   = blockIdx.y << 4;
    const int wave = threadIdx.x >> 5;
    const int m0   = ((blockIdx.z << 3) + wave) << 4;
    const int lane = threadIdx.x & 31;
    const int half = lane >> 4;     // 0: K%4 in {0,1}, 1: K%4 in {2,3}
    const int r    = lane & 15;

    const float* Xr = X + ((size_t)b * Nsz + n0 + r) * DIMsz; // row n0+r of X
    const float* Yr = Y + ((size_t)b * Msz + m0 + r) * DIMsz; // row m0+r of Y

    v8f acc = {};          // accumulates X@Y^T for this 16x16 tile
    float sx = 0.0f;       // partial sum of squares of row n0+r (half the Ks)
    float sy = 0.0f;       // partial sum of squares of row m0+r (half the Ks)

#pragma unroll
    for (int kk = 0; kk < DIMsz; kk += 4) {
        v2f a  = *(const v2f*)(Xr + kk + 2 * half);   // A[M=r, K=kk+2h .. +1]
        v2f bb = *(const v2f*)(Yr + kk + 2 * half);   // B[K, N=r] (Y row-major)
        sx += a.x * a.x + a.y * a.y;
        sy += bb.x * bb.x + bb.y * bb.y;
        acc = __builtin_amdgcn_wmma_f32_16x16x4_f32(
            /*neg_a=*/false, a, /*neg_b=*/false, bb,
            /*c_mod=*/(short)0, acc, /*reuse_a=*/false, /*reuse_b=*/false);
    }
    // Combine the two K-halves: after this, lane (h, r) holds full ||X_{n0+r}||^2
    // and ||Y_{m0+r}||^2 for both h=0 and h=1.
    sx += __shfl_xor(sx, 16, 32);
    sy += __shfl_xor(sy, 16, 32);

    const float y2c = sy;  // column written by this lane is m0+r
    float* Db = D + (size_t)b * Nsz * Msz;
#pragma unroll
    for (int v = 0; v < 8; ++v) {
        const int row = v + 8 * half;                 // D-tile row for acc[v]
        const float x2r = __shfl(sx, row, 32);        // ||X_{n0+row}||^2
        Db[(size_t)(n0 + row) * Msz + (m0 + r)] = x2r + y2c - 2.0f * acc[v];
    }
}

// --------------------------- Phase 2: DP sweep -----------------------------
// One block per batch. 512 threads; thread tid owns matrix row i = tid+1 of
// the (N+1)x(M+1) DP table. Three rolling diagonals live in LDS. One barrier
// per anti-diagonal: the buffer written at step k was last read (as d2) at
// step k-1, so a single s_barrier per step is sufficient.
__global__ __launch_bounds__(512)
void softdtw_dp(const float* __restrict__ D,
                const int* __restrict__ lengths,
                float* __restrict__ out) {
    const int b   = blockIdx.x;
    const int tid = threadIdx.x;       // 0..511
    const int i   = tid + 1;           // DP row handled by this thread
    const int len = lengths[b];

    __shared__ float buf[3][Nsz + 1];  // 3 * 513 floats = ~6 KB of 320 KB LDS

    // diag k=0: only cell (0,0)=0; diag k=1: all boundary (BIG)
    buf[0][i] = BIGF;
    buf[1][i] = BIGF;
    if (tid == 0) { buf[0][0] = 0.0f; buf[1][0] = BIGF; buf[2][0] = BIGF; }
    __syncthreads();

    const float* Drow = D + ((size_t)b * Nsz + tid) * Msz;  // D row (i-1)
    const int kout = len + Msz;        // diagonal carrying the answer

#pragma unroll 1
    for (int k = 2; k <= Nsz + Msz; ++k) {
        float* d2 = buf[(k - 2) % 3];  // diag k-2
        float* d1 = buf[(k - 1) % 3];  // diag k-1
        float* dn = buf[k % 3];        // diag k (being written)

        const int j = k - i;
        float nv = BIGF;
        if (j >= 1 && j <= Msz && i <= len) {
            const float a = d2[i - 1]; // R[i-1][j-1]
            const float p = d1[i - 1]; // R[i-1][j]
            const float c = d1[i];     // R[i][j-1]
            // softmin_{gamma=1}(a,p,c) = mn - log(sum exp(mn - .))
            const float mn = fminf(a, fminf(p, c));
            const float s  = __expf(mn - a) + __expf(mn - p) + __expf(mn - c);
            nv = Drow[j - 1] + (mn - __logf(s));
        }
        dn[i] = nv;
        if (tid == 0) dn[0] = BIGF;    // row 0 is always boundary for k>=1
        if (k == kout && i == len) out[b] = nv;  // R[len, M]
        __syncthreads();
    }
}

extern "C" void kernel_launch(void* const* d_in, const int* in_sizes, int n_in,
                              void* d_out, int out_size, void* d_ws, size_t ws_size,
                              hipStream_t stream) {
    const float* X        = (const float*)d_in[0];   // (B, N, DIM) f32
    const float* Y        = (const float*)d_in[1];   // (B, M, DIM) f32
    const int*   lengths  = (const int*)d_in[2];     // (B,) i32
    float*       out      = (float*)d_out;           // (B,) f32
    float*       D        = (float*)d_ws;            // B*N*M f32 = 64 MiB scratch

    dim3 grid1(Bsz, Nsz / 16, Msz / (16 * 8));       // (64, 32, 4)
    softdtw_pairdist_wmma<<<grid1, 256, 0, stream>>>(X, Y, D);

    softdtw_dp<<<dim3(Bsz), dim3(Nsz), 0, stream>>>(D, lengths, out);
}